// AttentionMergeMask_45913200394450
// MI455X (gfx1250) — compile-verified
//
#include <hip/hip_runtime.h>
#include <stdint.h>

// ---------------------------------------------------------------------------
// Problem: bs=32, ch=256, hw=1024.
//   scale[b,p] = mask/||fg[:,p]+eps||
//   att = K @ S ; final = att*(1-m)+fg*m ; out = W @ [fg;final] + bias
// Memory-bound: ~200MB traffic vs 26 GFLOP -> stream S once, keep `final`
// in LDS, overlap HBM with WMMA via async global->LDS double buffering.
// ---------------------------------------------------------------------------
#define EPS 1e-7f
constexpr int BS = 32;
constexpr int CH = 256;
constexpr int HW = 1024;

constexpr int NT = 64;    // positions per workgroup
constexpr int KT = 32;    // K chunk per pipeline stage

// LDS strides (floats):
//  AROW=36: A/W panels stored [row][k] -> A-fragment = contiguous b64;
//           36c mod 64 spans all multiples of 4 -> half-waves use disjoint banks.
//  BROW=72: 2*72 % 64 == 16 -> fragment half-waves hit disjoint bank ranges.
constexpr int AROW = 36;
constexpr int BROW = 72;

constexpr int A_BUF  = 256 * AROW * 4;   // 36864 B
constexpr int B_BUF  = KT  * BROW * 4;   //  9216 B
constexpr int FG_BUF = CH  * BROW * 4;   // 73728 B

constexpr int OFF_A0 = 0;
constexpr int OFF_A1 = A_BUF;
constexpr int OFF_F  = 2 * A_BUF;        // final tile; phase-1 S buffers alias here
constexpr int OFF_B0 = OFF_F;
constexpr int OFF_B1 = OFF_F + B_BUF;
constexpr int OFF_G  = OFF_F + FG_BUF;   // fg tile at output positions
constexpr int OFF_CS = OFF_G + FG_BUF;   // 4x64 column-sum partials
constexpr int SMEM_BYTES = OFF_CS + 4 * 64 * 4;   // 222208 B (< 320KB WGP)

typedef __attribute__((ext_vector_type(2))) float v2f;
typedef __attribute__((ext_vector_type(8))) float v8f;

// Low 32 bits of a generic pointer into LDS == LDS byte address (flat aperture).
__device__ __forceinline__ unsigned ldsa(const void* p) {
  return (unsigned)(uintptr_t)p;
}
// CDNA5 async DMA: 16B per lane, global -> LDS, tracked by ASYNCcnt.
__device__ __forceinline__ void async_cp16(unsigned lds, const float* g) {
  asm volatile("global_load_async_to_lds_b128 %0, %1, off"
               :: "v"(lds), "v"((unsigned long long)(uintptr_t)g)
               : "memory");
}

// ---------------------------------------------------------------------------
// Kernel 1: scale[b,p] = mask[b,p] * rsqrt( sum_c (fg[b,c,p]+eps)^2 )
// ---------------------------------------------------------------------------
__global__ void AMM_scale_kernel(const float* __restrict__ fg,
                                 const float* __restrict__ mask,
                                 float* __restrict__ scale) {
  int idx = blockIdx.x * blockDim.x + threadIdx.x;   // b*HW + p
  int b = idx >> 10;
  int p = idx & (HW - 1);
  const float* f = fg + (size_t)b * CH * HW + p;
  float s = 0.0f;
#pragma unroll 8
  for (int c = 0; c < CH; ++c) {
    float v = f[(size_t)c * HW] + EPS;
    s = fmaf(v, v, s);
  }
  scale[idx] = mask[idx] * rsqrtf(s);
}

// ---------------------------------------------------------------------------
// Kernel 2: fused GEMM1 -> mask mix -> GEMM2, async double-buffered panels.
// One WG per (batch, 64-position tile); 256 threads = 8 waves.
// Wave w owns M-tiles {2w,2w+1} x 4 N-tiles (8 v8f accumulators).
// ---------------------------------------------------------------------------
__global__ __launch_bounds__(256)
void AMM_fused_kernel(const float* __restrict__ fg,
                      const float* __restrict__ mask,
                      const float* __restrict__ S,
                      const float* __restrict__ W,
                      const float* __restrict__ bias,
                      const float* __restrict__ scale,
                      float* __restrict__ out) {
  __shared__ __attribute__((aligned(16))) unsigned char smem[SMEM_BYTES];
  float* A0 = (float*)(smem + OFF_A0);   // [256][AROW] fg panel / W panel
  float* A1 = (float*)(smem + OFF_A1);
  float* B0 = (float*)(smem + OFF_B0);   // [KT][BROW] S panel (phase 1 only)
  float* B1 = (float*)(smem + OFF_B1);
  float* F_lds = (float*)(smem + OFF_F); // [CH][BROW] final tile
  float* G_lds = (float*)(smem + OFF_G); // [CH][BROW] fg tile at out positions
  float* CS    = (float*)(smem + OFF_CS);// [4][64] colsum partials

  const int b    = blockIdx.y;
  const int n0   = blockIdx.x * NT;
  const int tid  = threadIdx.x;
  const int lane = tid & 31;
  const int wv   = tid >> 5;
  const int hi   = lane >> 4;
  const int lo   = lane & 15;

  const float* fgB = fg    + (size_t)b * CH * HW;
  const float* Sb  = S     + (size_t)b * HW * HW;
  const float* mb  = mask  + (size_t)b * HW;
  const float* scb = scale + (size_t)b * HW;

  // Async issue helpers: 8 (A) + 2 (B) b128 copies per thread per stage.
  auto issueA = [&](int kb, float* Ab) {
    const int c0  = tid >> 3;           // 0..31
    const int kk0 = (tid & 7) * 4;      // 0..28
#pragma unroll
    for (int i = 0; i < 8; ++i) {
      int c = c0 + 32 * i;
      async_cp16(ldsa(Ab + c * AROW + kk0), fgB + (size_t)c * HW + kb + kk0);
    }
  };
  auto issueB = [&](int kb, float* Bb) {
    const int nch = (tid & 15) * 4;     // 0..60
    const int kk  = tid >> 4;           // 0..15
#pragma unroll
    for (int i = 0; i < 2; ++i) {
      int k = kk + 16 * i;
      async_cp16(ldsa(Bb + k * BROW + nch), Sb + (size_t)(kb + k) * HW + n0 + nch);
    }
  };
  auto issueW = [&](int ib, float* Ab) {
    const int o0  = tid >> 3;
    const int kk0 = (tid & 7) * 4;
#pragma unroll
    for (int i = 0; i < 8; ++i) {
      int o = o0 + 32 * i;
      async_cp16(ldsa(Ab + o * AROW + kk0), W + (size_t)o * (2 * CH) + ib + kk0);
    }
  };

  // ----------------- Phase 1: att = (fg) @ (scale*S), eps folded out --------
  v8f acc[2][4];
#pragma unroll
  for (int m = 0; m < 2; ++m)
#pragma unroll
    for (int n = 0; n < 4; ++n)
      acc[m][n] = {};

  issueA(0, A0);
  issueB(0, B0);
  float csum = 0.0f;   // partial of sum_k scale[k]*S[k, mycol]

  for (int t = 0; t < HW / KT; ++t) {
    const int kb = t * KT;
    float* Ac = (t & 1) ? A1 : A0;
    float* Bc = (t & 1) ? B1 : B0;
    __syncthreads();                       // prior readers done with next bufs
    if (t + 1 < HW / KT) {
      issueA(kb + KT, (t & 1) ? A0 : A1);
      issueB(kb + KT, (t & 1) ? B0 : B1);
      asm volatile("s_wait_asynccnt 0xa" ::: "memory");  // current stage landed
    } else {
      asm volatile("s_wait_asynccnt 0x0" ::: "memory");
    }
    __syncthreads();                       // all waves' copies visible

    // Scale S panel in place: B'[k][n] = scale[kb+k]*S ; accumulate colsum.
    {
      int n  = tid & 63;
      int k0 = tid >> 6;                   // 0..3
#pragma unroll
      for (int i = 0; i < 8; ++i) {
        int kk = k0 + 4 * i;
        float v = Bc[kk * BROW + n] * scb[kb + kk];
        Bc[kk * BROW + n] = v;
        csum += v;
      }
    }
    __syncthreads();

#pragma unroll
    for (int kq = 0; kq < KT; kq += 4) {
      const int kk = kq + 2 * hi;
      v2f afr[2], bfr[4];
#pragma unroll
      for (int m = 0; m < 2; ++m) {
        int c = (2 * wv + m) * 16 + lo;
        afr[m] = *reinterpret_cast<const v2f*>(Ac + c * AROW + kk);  // b64
      }
#pragma unroll
      for (int n = 0; n < 4; ++n) {
        int col = n * 16 + lo;
        bfr[n].x = Bc[kk * BROW + col];
        bfr[n].y = Bc[(kk + 1) * BROW + col];
      }
#pragma unroll
      for (int m = 0; m < 2; ++m)
#pragma unroll
        for (int n = 0; n < 4; ++n)
          acc[m][n] = __builtin_amdgcn_wmma_f32_16x16x4_f32(
              false, afr[m], false, bfr[n], (short)0, acc[m][n], false, false);
    }
  }

  // Reduce eps-correction column sums (4 partials per column).
  __syncthreads();
  CS[(tid >> 6) * 64 + (tid & 63)] = csum;
  __syncthreads();

  // ----------------- Phase 2: att + eps*colsum, mask mix, stage F/G ---------
#pragma unroll
  for (int m = 0; m < 2; ++m) {
    int mt = 2 * wv + m;
#pragma unroll
    for (int n = 0; n < 4; ++n) {
      int col = n * 16 + lo;
      int p   = n0 + col;
      float corr = EPS * (CS[col] + CS[64 + col] + CS[128 + col] + CS[192 + col]);
      float mk   = mb[p];
#pragma unroll
      for (int r = 0; r < 8; ++r) {
        int c = mt * 16 + r + 8 * hi;
        float fgv = fgB[(size_t)c * HW + p];
        float att = acc[m][n][r] + corr;
        float fin = att * (1.0f - mk) + fgv * mk;
        F_lds[c * BROW + col] = fin;
        G_lds[c * BROW + col] = fgv;
      }
    }
  }

  // ----------------- Phase 3: out = W @ [fg;final] + bias -------------------
  v8f acc2[2][4];
#pragma unroll
  for (int m = 0; m < 2; ++m) {
#pragma unroll
    for (int r = 0; r < 8; ++r) {
      int o = (2 * wv + m) * 16 + r + 8 * hi;
      float bv = bias[o];
#pragma unroll
      for (int n = 0; n < 4; ++n)
        acc2[m][n][r] = bv;
    }
  }

  issueW(0, A0);   // A buffers free; F/G untouched by W panels
  for (int t = 0; t < (2 * CH) / KT; ++t) {
    const int ib = t * KT;
    float* Ac = (t & 1) ? A1 : A0;
    __syncthreads();                       // F/G visible (t==0); prev readers done
    if (t + 1 < (2 * CH) / KT) {
      issueW(ib + KT, (t & 1) ? A0 : A1);
      asm volatile("s_wait_asynccnt 0x8" ::: "memory");
    } else {
      asm volatile("s_wait_asynccnt 0x0" ::: "memory");
    }
    __syncthreads();

    const float* Bsrc = (ib < CH) ? (G_lds + (size_t)ib * BROW)
                                  : (F_lds + (size_t)(ib - CH) * BROW);
#pragma unroll
    for (int kq = 0; kq < KT; kq += 4) {
      const int kk = kq + 2 * hi;
      v2f afr[2], bfr[4];
#pragma unroll
      for (int m = 0; m < 2; ++m) {
        int o = (2 * wv + m) * 16 + lo;
        afr[m] = *reinterpret_cast<const v2f*>(Ac + o * AROW + kk);
      }
#pragma unroll
      for (int n = 0; n < 4; ++n) {
        int col = n * 16 + lo;
        bfr[n].x = Bsrc[kk * BROW + col];
        bfr[n].y = Bsrc[(kk + 1) * BROW + col];
      }
#pragma unroll
      for (int m = 0; m < 2; ++m)
#pragma unroll
        for (int n = 0; n < 4; ++n)
          acc2[m][n] = __builtin_amdgcn_wmma_f32_16x16x4_f32(
              false, afr[m], false, bfr[n], (short)0, acc2[m][n], false, false);
    }
  }

  // ----------------- Epilogue ------------------------------------------------
  float* ob = out + (size_t)b * CH * HW;
#pragma unroll
  for (int m = 0; m < 2; ++m) {
#pragma unroll
    for (int n = 0; n < 4; ++n) {
      int p = n0 + n * 16 + lo;
#pragma unroll
      for (int r = 0; r < 8; ++r) {
        int o = (2 * wv + m) * 16 + r + 8 * hi;
        ob[(size_t)o * HW + p] = acc2[m][n][r];
      }
    }
  }
}

// ---------------------------------------------------------------------------
// Launch: inputs = {foreground, mask, attention_scores, comb_w, comb_b}
// d_ws: 32*1024 floats (128 KB) for scale[b,p].
// ---------------------------------------------------------------------------
extern "C" void kernel_launch(void* const* d_in, const int* in_sizes, int n_in,
                              void* d_out, int out_size, void* d_ws, size_t ws_size,
                              hipStream_t stream) {
  const float* fg   = (const float*)d_in[0];
  const float* mask = (const float*)d_in[1];
  const float* S    = (const float*)d_in[2];
  const float* W    = (const float*)d_in[3];
  const float* bias = (const float*)d_in[4];
  float* out   = (float*)d_out;
  float* scale = (float*)d_ws;

  AMM_scale_kernel<<<(BS * HW) / 256, 256, 0, stream>>>(fg, mask, scale);

  dim3 grid(HW / NT, BS);
  AMM_fused_kernel<<<grid, 256, 0, stream>>>(fg, mask, S, W, bias, scale, out);
}